// Net_53455162966108
// MI455X (gfx1250) — compile-verified
//
#include <hip/hip_runtime.h>
#include <cstdint>
#include <cstddef>

// ---- problem geometry -------------------------------------------------------
constexpr int IH = 1024;
constexpr int IW = 1024;
constexpr int HWPX = IH * IW;        // 1,048,576
constexpr int NBATCH = 16;
constexpr int NPLANES = 48;          // 16 batches * 3 channels

// unnormalized gaussian(5, std=1): exp(-0.5*(i-2)^2)
#define G0 0.1353352832366127f   // exp(-2)
#define G1 0.6065306597126334f   // exp(-0.5)

// ---- CDNA5 async global->LDS path (guarded; host pass sees fallback) --------
#if defined(__gfx1250__) && __has_builtin(__builtin_amdgcn_global_load_async_to_lds_b128)
#define HAVE_ASYNC_LDS 1
#else
#define HAVE_ASYNC_LDS 0
#endif

#if HAVE_ASYNC_LDS
// Parameter types per hipcc diagnostic: (v4i AS1*, v4i AS3*, Ii offset, Ii cpol)
typedef int v4i_t __attribute__((ext_vector_type(4)));
typedef __attribute__((address_space(1))) v4i_t* g_v4i_ptr;
typedef __attribute__((address_space(3))) v4i_t* l_v4i_ptr;
#endif

__device__ __forceinline__ void wait_async0() {
#if defined(__gfx1250__)
#if __has_builtin(__builtin_amdgcn_s_wait_asynccnt)
  __builtin_amdgcn_s_wait_asynccnt(0);
#else
  asm volatile("s_wait_asynccnt 0" ::: "memory");
#endif
#endif
}

// ============================================================================
// Fully fused Canny pipeline. One block = one (batch, 32x32 tile).
//
// LDS pool (floats), two overlaid regions with barriers between stages:
//   A  [0,    4800) : raw 3ch x 40x40 (halo 4)  ->  blurred 3ch x 36x36 (halo 2)
//   Bb [4800, 9120) : hb  3ch x 40x36           ->  mag 34x34 (halo 1)
// ============================================================================
extern "C" __global__ void __launch_bounds__(256)
k_canny(const float* __restrict__ img,
        const float* __restrict__ p_bh,  const float* __restrict__ p_bv,
        const float* __restrict__ p_bsh, const float* __restrict__ p_bsv,
        const float* __restrict__ p_d0,  const float* __restrict__ p_d1,
        float* __restrict__ o_blur, float* __restrict__ o_mag,
        float* __restrict__ o_ori,  float* __restrict__ o_thin,
        float* __restrict__ o_thr,  float* __restrict__ o_early) {
  __shared__ __align__(16) float pool[9120];
  float* Apool = pool;          // raw, later blurred
  float* Bpool = pool + 4800;   // hb, later mag

  const int tid  = threadIdx.x;
  const int bid  = blockIdx.x;
  const int b    = bid >> 10;            // 16 batches
  const int tile = bid & 1023;           // 32x32 tiles of 1024x1024
  const int ty = tile >> 5, tx = tile & 31;
  const int y0 = ty << 5,  x0 = tx << 5;
  const float bh  = *p_bh,  bv  = *p_bv;
  const float bsh = *p_bsh, bsv = *p_bsv;

  // ---- Stage 1: raw tile (3ch x 40x40, origin (y0-4, x0-4)) into LDS -------
  const bool interior = (tx >= 1) & (tx <= 30) & (ty >= 1) & (ty <= 30);
  (void)interior;
#if HAVE_ASYNC_LDS
  if (interior) {
    for (int c = 0; c < 3; ++c) {
      const float* gb = img + (size_t)(b * 3 + c) * HWPX
                      + (size_t)(y0 - 4) * IW + (x0 - 4);   // 16B aligned
      float* ld = Apool + c * 1600;
      for (int idx = tid; idx < 400; idx += 256) {          // 40 rows x 10 b128
        const int row = idx / 10;
        const int chunk = idx - row * 10;
        __builtin_amdgcn_global_load_async_to_lds_b128(
            (g_v4i_ptr)(gb + row * IW + chunk * 4),
            (l_v4i_ptr)(ld + row * 40 + chunk * 4), 0, 0);
      }
    }
    wait_async0();
  } else
#endif
  {
    for (int c = 0; c < 3; ++c) {
      const size_t pb = (size_t)(b * 3 + c) * HWPX;
      float* ld = Apool + c * 1600;
      for (int idx = tid; idx < 1600; idx += 256) {
        const int row = idx / 40;
        const int col = idx - row * 40;
        const int gy = y0 - 4 + row, gx = x0 - 4 + col;
        float v = 0.f;
        if (gy >= 0 && gy < IH && gx >= 0 && gx < IW)
          v = img[pb + (size_t)gy * IW + gx];
        ld[idx] = v;
      }
    }
  }
  __syncthreads();

  // ---- Stage 2: horizontal blur: hb[c][40][36], col cx ~ global x0-2+cx ----
  for (int c = 0; c < 3; ++c) {
    const float* rawc = Apool + c * 1600;
    float* hbc = Bpool + c * 1440;
    for (int idx = tid; idx < 1440; idx += 256) {
      const int row = idx / 36;
      const int cx = idx - row * 36;
      const int gy = y0 - 4 + row;
      float v = 0.f;
      if (gy >= 0 && gy < IH) {           // rows outside image: exact 0 pad
        const float* r = rawc + row * 40 + cx;   // raw col of (x0-2+cx+j-2)=cx+j
        v = bh + G0 * (r[0] + r[4]) + G1 * (r[1] + r[3]) + r[2];
      }
      hbc[idx] = v;
    }
  }
  __syncthreads();

  // ---- Stage 3: vertical blur -> blurred[c][36][36] (zero outside image),
  //      NT-store center 32x32 to o_blur ------------------------------------
  for (int c = 0; c < 3; ++c) {
    const float* hbc = Bpool + c * 1440;
    float* blc = Apool + c * 1296;        // overwrites dead raw
    for (int idx = tid; idx < 1296; idx += 256) {
      const int r = idx / 36;
      const int cx = idx - r * 36;
      const int gy = y0 - 2 + r, gx = x0 - 2 + cx;
      float v = 0.f;
      if (gy >= 0 && gy < IH && gx >= 0 && gx < IW) {
        const float* h = hbc + r * 36 + cx;      // hb row (r+i) ~ gy+i-2
        v = bv + G0 * (h[0] + h[4 * 36]) + G1 * (h[36] + h[3 * 36]) + h[2 * 36];
      }
      blc[idx] = v;
      if (r >= 2 && r < 34 && cx >= 2 && cx < 34) {
        const size_t pix = (size_t)(b * 3 + c) * HWPX + (size_t)gy * IW + gx;
        __builtin_nontemporal_store(v, &o_blur[pix]);
      }
    }
  }
  __syncthreads();

  // ---- Stage 4: Sobel magnitude on 34x34 (halo 1); center -> mag/early -----
  for (int idx = tid; idx < 1156; idx += 256) {
    const int m = idx / 34;
    const int n = idx - m * 34;           // mag(m,n) ~ global (y0-1+m, x0-1+n)
    float mag = 0.f;
#pragma unroll
    for (int c = 0; c < 3; ++c) {
      const float* p = Apool + c * 1296 + m * 36 + n;  // p[0] ~ (y0-2+m, x0-2+n)
      const float a00 = p[0],  a01 = p[1],  a02 = p[2];
      const float a10 = p[36],              a12 = p[38];
      const float a20 = p[72], a21 = p[73], a22 = p[74];
      const float gxc = (a00 - a02) + 2.f * (a10 - a12) + (a20 - a22) + bsh;
      const float gyc = (a00 - a20) + 2.f * (a01 - a21) + (a02 - a22) + bsv;
      mag += sqrtf(gxc * gxc + gyc * gyc);
    }
    Bpool[idx] = mag;                     // overwrites dead hb
    if (m >= 1 && m < 33 && n >= 1 && n < 33) {
      const size_t pix = (size_t)b * HWPX + (size_t)(y0 - 1 + m) * IW + (x0 - 1 + n);
      __builtin_nontemporal_store(mag, &o_mag[pix]);
      __builtin_nontemporal_store(mag < 10.f ? 0.f : mag, &o_early[pix]);
    }
  }
  __syncthreads();

  // ---- Stage 5: orientation + directional NMS + threshold on center 32x32 --
  // Per pixel exactly one of the 8 difference filters is selected by
  // k = orientation/45 in 0..4:  0->(0,+1) 1->(+1,+1) 2->(+1,0) 3->(+1,-1) 4->(0,-1)
  for (int idx = tid; idx < 1024; idx += 256) {
    const int lx = idx & 31, ly = idx >> 5;
    const int m = ly + 1, n = lx + 1;     // mag-tile coords
    float sx = 0.f, sy = 0.f;
#pragma unroll
    for (int c = 0; c < 3; ++c) {
      const float* p = Apool + c * 1296 + m * 36 + n;
      const float a00 = p[0],  a01 = p[1],  a02 = p[2];
      const float a10 = p[36],              a12 = p[38];
      const float a20 = p[72], a21 = p[73], a22 = p[74];
      sx += (a00 - a02) + 2.f * (a10 - a12) + (a20 - a22) + bsh;
      sy += (a00 - a20) + 2.f * (a01 - a21) + (a02 - a22) + bsv;
    }
    float go = atan2f(sy, sx) * (180.0f / 3.14159f);   // reference PI
    if (go < 0.f) go += 360.f;
    go = fmodf(go, 180.f);
    const float ori = rintf(go * (1.f / 45.f)) * 45.f; // {0,45,90,135,180}
    const int k = (int)(ori * (1.f / 45.f) + 0.5f);

    const int dy = (k >= 1 && k <= 3) ? 1 : 0;
    const int dx = (k <= 1) ? 1 : ((k >= 3) ? -1 : 0);
    const int y = y0 + ly, x = x0 + lx;
    const int ny = y + dy, nx = x + dx;

    float nb = 0.f;                                    // zero-pad at image border
    if (ny >= 0 && ny < IH && nx >= 0 && nx < IW)
      nb = Bpool[(m + dy) * 34 + (n + dx)];

    const float mag = Bpool[m * 34 + n];
    const float bias = (k < 4) ? *p_d0 : *p_d1;
    const float thin = (mag - nb + bias > 0.f) ? mag : 0.f;

    const size_t pix = (size_t)b * HWPX + (size_t)y * IW + x;
    __builtin_nontemporal_store(ori, &o_ori[pix]);
    __builtin_nontemporal_store(thin, &o_thin[pix]);
    __builtin_nontemporal_store((thin < 10.f) ? 0.f : thin, &o_thr[pix]);
  }
}

// ============================================================================
// Launcher. Output layout (flat concat, all f32):
//   [0)      blurred_img : 48*HW   (flat layout identical to [B,C,H,W])
//   [48HW)   grad_mag    : 16*HW
//   [64HW)   grad_orient : 16*HW
//   [80HW)   thin_edges  : 16*HW
//   [96HW)   thresholded : 16*HW
//   [112HW)  early_thresh: 16*HW
// ============================================================================
extern "C" void kernel_launch(void* const* d_in, const int* in_sizes, int n_in,
                              void* d_out, int out_size, void* d_ws, size_t ws_size,
                              hipStream_t stream) {
  const float* img = (const float*)d_in[0];
  const float* bh  = (const float*)d_in[1];
  const float* bv  = (const float*)d_in[2];
  const float* bsh = (const float*)d_in[3];
  const float* bsv = (const float*)d_in[4];
  const float* bd0 = (const float*)d_in[5];
  const float* bd1 = (const float*)d_in[6];
  (void)in_sizes; (void)n_in; (void)out_size; (void)d_ws; (void)ws_size;

  float* out     = (float*)d_out;
  float* o_blur  = out;
  float* o_mag   = out + (size_t)NPLANES * HWPX;
  float* o_ori   = o_mag  + (size_t)NBATCH * HWPX;
  float* o_thin  = o_ori  + (size_t)NBATCH * HWPX;
  float* o_thr   = o_thin + (size_t)NBATCH * HWPX;
  float* o_early = o_thr  + (size_t)NBATCH * HWPX;

  k_canny<<<NBATCH * 1024, 256, 0, stream>>>(img, bh, bv, bsh, bsv, bd0, bd1,
                                             o_blur, o_mag, o_ori, o_thin,
                                             o_thr, o_early);
}